// TreeMamba_90383291777361
// MI455X (gfx1250) — compile-verified
//
#include <hip/hip_runtime.h>
#include <math.h>

typedef __attribute__((ext_vector_type(2))) float v2f;
typedef __attribute__((ext_vector_type(4))) float v4f;
typedef __attribute__((ext_vector_type(8))) float v8f;

#define NB 8
#define NDEPTH 12
#define NL 4095
#define NDM 768
#define NDS 16
#define NDI 1536
#define NDTR 48
#define NKC 4
#define NXDB 80   // DT_RANK + 2*D_STATE

static __device__ __forceinline__ v2f ld2(const float* p) { return *(const v2f*)p; }
static __device__ __forceinline__ v4f ld4(const float* p) { return *(const v4f*)p; }
static __device__ __forceinline__ void st4(float* p, v4f v) { *(v4f*)p = v; }

// D = A(16x4, f32) * B(4x16, f32) + C(16x16, f32), wave32 WMMA
static __device__ __forceinline__ v8f wmma4(v2f a, v2f b, v8f c) {
  return __builtin_amdgcn_wmma_f32_16x16x4_f32(false, a, false, b, (short)0, c, false, false);
}

static __device__ __forceinline__ float sigmoidf_(float x) { return 1.f / (1.f + __expf(-x)); }
static __device__ __forceinline__ float softplusf_(float x) { return (x > 20.f) ? x : log1pf(__expf(x)); }

// ---------------- K0: Aneg = -exp(A_log) ----------------
__global__ void k_prep(const float* __restrict__ alog, float* __restrict__ an) {
  int i = blockIdx.x * 256 + threadIdx.x;
  if (i < NDI * NDS) an[i] = -__expf(alog[i]);
}

// ---------------- K1: in_proj GEMM ----------------
// C[e,l] = sum_d W[e,d] * x[b,l,d];  e<NDI -> XC[b,l,e] ; else Z[b,l,e-NDI]
// wave tile: 64 (e) x 32 (l), software-pipelined
__global__ __launch_bounds__(128) void k_inproj(const float* __restrict__ x,
                                                const float* __restrict__ W,
                                                float* __restrict__ XC,
                                                float* __restrict__ Zb) {
  const int lane = threadIdx.x & 31;
  const int wave = threadIdx.x >> 5;
  const int half = lane >> 4, lr = lane & 15;
  const int b  = blockIdx.z;
  const int l0 = blockIdx.x * 32;
  const int e0 = blockIdx.y * 256 + wave * 64;
  const int lA = l0 + lr;
  const int lB = l0 + 16 + lr;
  const int lcA = (lA < NL) ? lA : (NL - 1);
  const int lcB = (lB < NL) ? lB : (NL - 1);

  const float* bpA = x + ((size_t)(b * NL + lcA)) * NDM + 2 * half;
  const float* bpB = x + ((size_t)(b * NL + lcB)) * NDM + 2 * half;
  const float* ap0 = W + (size_t)(e0 +  0 + lr) * NDM + 2 * half;
  const float* ap1 = W + (size_t)(e0 + 16 + lr) * NDM + 2 * half;
  const float* ap2 = W + (size_t)(e0 + 32 + lr) * NDM + 2 * half;
  const float* ap3 = W + (size_t)(e0 + 48 + lr) * NDM + 2 * half;

  v8f c00 = {}, c01 = {}, c10 = {}, c11 = {};
  v8f c20 = {}, c21 = {}, c30 = {}, c31 = {};

  v2f fA = ld2(bpA), fB = ld2(bpB);
  v2f f0 = ld2(ap0), f1 = ld2(ap1), f2 = ld2(ap2), f3 = ld2(ap3);
#pragma unroll 2
  for (int k = 4; k < NDM; k += 4) {
    v2f nA = ld2(bpA + k), nB = ld2(bpB + k);
    v2f n0 = ld2(ap0 + k), n1 = ld2(ap1 + k);
    v2f n2 = ld2(ap2 + k), n3 = ld2(ap3 + k);
    c00 = wmma4(f0, fA, c00); c01 = wmma4(f0, fB, c01);
    c10 = wmma4(f1, fA, c10); c11 = wmma4(f1, fB, c11);
    c20 = wmma4(f2, fA, c20); c21 = wmma4(f2, fB, c21);
    c30 = wmma4(f3, fA, c30); c31 = wmma4(f3, fB, c31);
    fA = nA; fB = nB; f0 = n0; f1 = n1; f2 = n2; f3 = n3;
  }
  c00 = wmma4(f0, fA, c00); c01 = wmma4(f0, fB, c01);
  c10 = wmma4(f1, fA, c10); c11 = wmma4(f1, fB, c11);
  c20 = wmma4(f2, fA, c20); c21 = wmma4(f2, fB, c21);
  c30 = wmma4(f3, fA, c30); c31 = wmma4(f3, fB, c31);

#define STORE_SUB(ACC, T, ROWO) do {                                   \
    int e = e0 + (T) * 16 + half * 8;                                  \
    float* dst = (e < NDI) ? (XC + (ROWO) + e) : (Zb + (ROWO) + (e - NDI)); \
    v4f lo = { ACC[0], ACC[1], ACC[2], ACC[3] };                       \
    v4f hi = { ACC[4], ACC[5], ACC[6], ACC[7] };                       \
    st4(dst, lo); st4(dst + 4, hi);                                    \
  } while (0)
  if (lA < NL) {
    const size_t ro = (size_t)(b * NL + lA) * NDI;
    STORE_SUB(c00, 0, ro); STORE_SUB(c10, 1, ro);
    STORE_SUB(c20, 2, ro); STORE_SUB(c30, 3, ro);
  }
  if (lB < NL) {
    const size_t ro = (size_t)(b * NL + lB) * NDI;
    STORE_SUB(c01, 0, ro); STORE_SUB(c11, 1, ro);
    STORE_SUB(c21, 2, ro); STORE_SUB(c31, 3, ro);
  }
#undef STORE_SUB
}

// ---------------- K2: gathered conv + SiLU ----------------
__global__ __launch_bounds__(384) void k_conv(const float* __restrict__ XC,
                                              const int* __restrict__ ci,
                                              const float* __restrict__ cw,
                                              const float* __restrict__ cb,
                                              float* __restrict__ XA) {
  const int bl = blockIdx.x;
  const int b = bl / NL, l = bl - b * NL;
  const int d0 = threadIdx.x * 4;
  const int base = b * (NKC * NL) + l * NKC;
  const float* xcb = XC + (size_t)b * NL * NDI;

  float a0 = cb[d0 + 0], a1 = cb[d0 + 1], a2 = cb[d0 + 2], a3 = cb[d0 + 3];
#pragma unroll
  for (int k = 0; k < NKC; ++k) {
    int idx = ci[base + k];
    if (idx > 0) {
      v4f v = ld4(xcb + (size_t)(idx - 1) * NDI + d0);
      a0 += v[0] * cw[(d0 + 0) * NKC + k];
      a1 += v[1] * cw[(d0 + 1) * NKC + k];
      a2 += v[2] * cw[(d0 + 2) * NKC + k];
      a3 += v[3] * cw[(d0 + 3) * NKC + k];
    }
  }
  v4f o = { a0 * sigmoidf_(a0), a1 * sigmoidf_(a1), a2 * sigmoidf_(a2), a3 * sigmoidf_(a3) };
  st4(XA + (size_t)(b * NL + l) * NDI + d0, o);
}

// ---------------- K3: x_db = xl @ x_proj_w.T  (M=P, N=80, K=1536) ----------------
__global__ __launch_bounds__(128) void k_xdb(const float* __restrict__ XA,
                                             const int* __restrict__ idxa,
                                             const float* __restrict__ xpw,
                                             float* __restrict__ XDB,
                                             int P, int levoff) {
  const int lane = threadIdx.x & 31;
  const int wave = threadIdx.x >> 5;
  const int half = lane >> 4, lr = lane & 15;
  const int p0 = blockIdx.x * 64 + wave * 16;
  const int pa = p0 + lr;
  const int pc = (pa < P) ? pa : (P - 1);
  const int bi = idxa[2 * (levoff + pc)];
  const int li = idxa[2 * (levoff + pc) + 1];

  const float* arow = XA + ((size_t)bi * NL + li) * NDI + 2 * half;
  const float* b0 = xpw + (size_t)( 0 + lr) * NDI + 2 * half;
  const float* b1 = xpw + (size_t)(16 + lr) * NDI + 2 * half;
  const float* b2 = xpw + (size_t)(32 + lr) * NDI + 2 * half;
  const float* b3 = xpw + (size_t)(48 + lr) * NDI + 2 * half;
  const float* b4 = xpw + (size_t)(64 + lr) * NDI + 2 * half;

  v8f c0 = {}, c1 = {}, c2 = {}, c3 = {}, c4 = {};
  v2f fa = ld2(arow);
  v2f g0 = ld2(b0), g1 = ld2(b1), g2 = ld2(b2), g3 = ld2(b3), g4 = ld2(b4);
#pragma unroll 2
  for (int k = 4; k < NDI; k += 4) {
    v2f na = ld2(arow + k);
    v2f h0 = ld2(b0 + k), h1 = ld2(b1 + k), h2 = ld2(b2 + k);
    v2f h3 = ld2(b3 + k), h4 = ld2(b4 + k);
    c0 = wmma4(fa, g0, c0);
    c1 = wmma4(fa, g1, c1);
    c2 = wmma4(fa, g2, c2);
    c3 = wmma4(fa, g3, c3);
    c4 = wmma4(fa, g4, c4);
    fa = na; g0 = h0; g1 = h1; g2 = h2; g3 = h3; g4 = h4;
  }
  c0 = wmma4(fa, g0, c0);
  c1 = wmma4(fa, g1, c1);
  c2 = wmma4(fa, g2, c2);
  c3 = wmma4(fa, g3, c3);
  c4 = wmma4(fa, g4, c4);

  const int pb = p0 + half * 8;
#pragma unroll
  for (int r = 0; r < 8; ++r) {
    int p = pb + r;
    if (p < P) {
      float* row = XDB + (size_t)p * NXDB + lr;
      row[0]  = c0[r]; row[16] = c1[r]; row[32] = c2[r]; row[48] = c3[r]; row[64] = c4[r];
    }
  }
}

// ---------------- K4: dt = softplus(x_db[:, :48] @ dt_proj_w.T + b)  (N=1536, K=48) ----------------
__global__ __launch_bounds__(128) void k_dt(const float* __restrict__ XDB,
                                            const float* __restrict__ dpw,
                                            const float* __restrict__ dpb,
                                            float* __restrict__ DTb,
                                            int P) {
  const int lane = threadIdx.x & 31;
  const int wave = threadIdx.x >> 5;
  const int half = lane >> 4, lr = lane & 15;
  const int n0 = (blockIdx.x * 4 + wave) * 64;
  const int p0 = blockIdx.y * 16;
  const int pa = p0 + lr;
  const int pc = (pa < P) ? pa : (P - 1);

  const float* arow = XDB + (size_t)pc * NXDB + 2 * half;
  const float* b0 = dpw + (size_t)(n0 +  0 + lr) * NDTR + 2 * half;
  const float* b1 = dpw + (size_t)(n0 + 16 + lr) * NDTR + 2 * half;
  const float* b2 = dpw + (size_t)(n0 + 32 + lr) * NDTR + 2 * half;
  const float* b3 = dpw + (size_t)(n0 + 48 + lr) * NDTR + 2 * half;

  v8f c0 = {}, c1 = {}, c2 = {}, c3 = {};
  v2f fa = ld2(arow);
  v2f g0 = ld2(b0), g1 = ld2(b1), g2 = ld2(b2), g3 = ld2(b3);
#pragma unroll
  for (int k = 4; k < NDTR; k += 4) {
    v2f na = ld2(arow + k);
    v2f h0 = ld2(b0 + k), h1 = ld2(b1 + k), h2 = ld2(b2 + k), h3 = ld2(b3 + k);
    c0 = wmma4(fa, g0, c0);
    c1 = wmma4(fa, g1, c1);
    c2 = wmma4(fa, g2, c2);
    c3 = wmma4(fa, g3, c3);
    fa = na; g0 = h0; g1 = h1; g2 = h2; g3 = h3;
  }
  c0 = wmma4(fa, g0, c0);
  c1 = wmma4(fa, g1, c1);
  c2 = wmma4(fa, g2, c2);
  c3 = wmma4(fa, g3, c3);

  const int pb = p0 + half * 8;
#pragma unroll
  for (int r = 0; r < 8; ++r) {
    int p = pb + r;
    if (p < P) {
      float* drow = DTb + (size_t)p * NDI;
      int n = n0 + lr;
      drow[n +  0] = softplusf_(c0[r] + dpb[n +  0]);
      drow[n + 16] = softplusf_(c1[r] + dpb[n + 16]);
      drow[n + 32] = softplusf_(c2[r] + dpb[n + 32]);
      drow[n + 48] = softplusf_(c3[r] + dpb[n + 48]);
    }
  }
}

// ---------------- K5: fused parent-blend + SSM update + y ----------------
__global__ __launch_bounds__(256) void k_ssm(const float* __restrict__ XA,
                                             const float* __restrict__ Zb,
                                             const int* __restrict__ idxa,
                                             const int* __restrict__ sta,
                                             const float* __restrict__ XDB,
                                             const float* __restrict__ DTb,
                                             const float* __restrict__ AN,
                                             const float* __restrict__ lavg,
                                             const float* __restrict__ Dpar,
                                             const float* __restrict__ prev,
                                             float* __restrict__ cur,
                                             float* __restrict__ Yb,
                                             int P, int levoff, int stoff, int leaf) {
  const int p = blockIdx.y;
  const int d = blockIdx.x * 256 + threadIdx.x;
  const int bi = idxa[2 * (levoff + p)];
  const int li = idxa[2 * (levoff + p) + 1];
  const size_t ro = ((size_t)bi * NL + li) * NDI + d;
  const float xv = XA[ro];
  const float zv = Zb[ro];
  const float dt = DTb[(size_t)p * NDI + d];

  const float* xdbp = XDB + (size_t)p * NXDB;
  v4f Bm[4], Cm[4];
#pragma unroll
  for (int q = 0; q < 4; ++q) {
    Bm[q] = ld4(xdbp + NDTR + 4 * q);
    Cm[q] = ld4(xdbp + NDTR + NDS + 4 * q);
  }

  v4f ssin[4];
  if (leaf) {
#pragma unroll
    for (int q = 0; q < 4; ++q) ssin[q] = (v4f){0.f, 0.f, 0.f, 0.f};
  } else {
    const int s0 = sta[2 * (stoff + p)];
    const int s1 = sta[2 * (stoff + p) + 1];
    const float* lp = lavg + (size_t)d * NDS;
    const float* pl = prev + ((size_t)(s0 - 1) * NDI + d) * NDS;
    const float* pr = prev + ((size_t)(s1 - 1) * NDI + d) * NDS;
#pragma unroll
    for (int q = 0; q < 4; ++q) {
      v4f la = ld4(lp + 4 * q);
      v4f lv = (s0 > 0) ? ld4(pl + 4 * q) : (v4f){0.f, 0.f, 0.f, 0.f};
      v4f rv = (s1 > 0) ? ld4(pr + 4 * q) : (v4f){0.f, 0.f, 0.f, 0.f};
      ssin[q] = la * lv + ((v4f){1.f, 1.f, 1.f, 1.f} - la) * rv;
    }
  }

  const float dtx = dt * xv;
  const float* an = AN + (size_t)d * NDS;
  float* so = cur + ((size_t)p * NDI + d) * NDS;
  float yacc = 0.f;
#pragma unroll
  for (int q = 0; q < 4; ++q) {
    v4f a = ld4(an + 4 * q);
    v4f o;
#pragma unroll
    for (int j = 0; j < 4; ++j) {
      float dA = __expf(dt * a[j]);
      float v = ssin[q][j] * dA + dtx * Bm[q][j];
      o[j] = v;
      yacc += v * Cm[q][j];
    }
    st4(so + 4 * q, o);
  }
  float y = yacc + Dpar[d] * xv;
  y *= zv * sigmoidf_(zv);
  Yb[(size_t)p * NDI + d] = y;
}

// ---------------- K6: out GEMM + scatter  (M=P, N=768, K=1536) ----------------
// wave tile: 32 (p) x 64 (n), software-pipelined
__global__ __launch_bounds__(128) void k_out(const float* __restrict__ Yb,
                                             const float* __restrict__ opw,
                                             const int* __restrict__ idxa,
                                             float* __restrict__ out,
                                             int P, int levoff) {
  const int lane = threadIdx.x & 31;
  const int wave = threadIdx.x >> 5;
  const int half = lane >> 4, lr = lane & 15;
  const int n0 = (blockIdx.x * 4 + wave) * 64;
  const int p0 = blockIdx.y * 32;
  const int paA = p0 + lr;
  const int paB = p0 + 16 + lr;
  const int pcA = (paA < P) ? paA : (P - 1);
  const int pcB = (paB < P) ? paB : (P - 1);

  const float* arA = Yb + (size_t)pcA * NDI + 2 * half;
  const float* arB = Yb + (size_t)pcB * NDI + 2 * half;
  const float* b0 = opw + (size_t)(n0 +  0 + lr) * NDI + 2 * half;
  const float* b1 = opw + (size_t)(n0 + 16 + lr) * NDI + 2 * half;
  const float* b2 = opw + (size_t)(n0 + 32 + lr) * NDI + 2 * half;
  const float* b3 = opw + (size_t)(n0 + 48 + lr) * NDI + 2 * half;

  v8f cA0 = {}, cA1 = {}, cA2 = {}, cA3 = {};
  v8f cB0 = {}, cB1 = {}, cB2 = {}, cB3 = {};
  v2f fA = ld2(arA), fB = ld2(arB);
  v2f g0 = ld2(b0), g1 = ld2(b1), g2 = ld2(b2), g3 = ld2(b3);
#pragma unroll 2
  for (int k = 4; k < NDI; k += 4) {
    v2f nA = ld2(arA + k), nB = ld2(arB + k);
    v2f h0 = ld2(b0 + k), h1 = ld2(b1 + k), h2 = ld2(b2 + k), h3 = ld2(b3 + k);
    cA0 = wmma4(fA, g0, cA0); cB0 = wmma4(fB, g0, cB0);
    cA1 = wmma4(fA, g1, cA1); cB1 = wmma4(fB, g1, cB1);
    cA2 = wmma4(fA, g2, cA2); cB2 = wmma4(fB, g2, cB2);
    cA3 = wmma4(fA, g3, cA3); cB3 = wmma4(fB, g3, cB3);
    fA = nA; fB = nB; g0 = h0; g1 = h1; g2 = h2; g3 = h3;
  }
  cA0 = wmma4(fA, g0, cA0); cB0 = wmma4(fB, g0, cB0);
  cA1 = wmma4(fA, g1, cA1); cB1 = wmma4(fB, g1, cB1);
  cA2 = wmma4(fA, g2, cA2); cB2 = wmma4(fB, g2, cB2);
  cA3 = wmma4(fA, g3, cA3); cB3 = wmma4(fB, g3, cB3);

#define STORE_ROWS(C0, C1, C2, C3, PB) do {                         \
    size_t rb[8];                                                   \
    _Pragma("unroll")                                               \
    for (int r = 0; r < 8; ++r) {                                   \
      int p = ((PB) + r < P) ? ((PB) + r) : (P - 1);                \
      int bi_ = idxa[2 * (levoff + p)];                             \
      int li_ = idxa[2 * (levoff + p) + 1];                         \
      rb[r] = ((size_t)bi_ * NL + li_) * NDM;                       \
    }                                                               \
    _Pragma("unroll")                                               \
    for (int r = 0; r < 8; ++r) {                                   \
      if ((PB) + r < P) {                                           \
        float* dst = out + rb[r] + n0 + lr;                         \
        dst[0] = C0[r]; dst[16] = C1[r]; dst[32] = C2[r]; dst[48] = C3[r]; \
      }                                                             \
    }                                                               \
  } while (0)
  STORE_ROWS(cA0, cA1, cA2, cA3, p0 + half * 8);
  STORE_ROWS(cB0, cB1, cB2, cB3, p0 + 16 + half * 8);
#undef STORE_ROWS
}

// ---------------- host ----------------
extern "C" void kernel_launch(void* const* d_in, const int* in_sizes, int n_in,
                              void* d_out, int out_size, void* d_ws, size_t ws_size,
                              hipStream_t stream) {
  (void)in_sizes; (void)n_in; (void)out_size; (void)ws_size;
  const float* x    = (const float*)d_in[0];
  const int*   ci   = (const int*)d_in[1];
  const int*   idxa = (const int*)d_in[2];
  const int*   sta  = (const int*)d_in[3];
  const float* ipw  = (const float*)d_in[4];
  const float* cw   = (const float*)d_in[5];
  const float* cb   = (const float*)d_in[6];
  const float* xpw  = (const float*)d_in[7];
  const float* dpw  = (const float*)d_in[8];
  const float* dpb  = (const float*)d_in[9];
  const float* alog = (const float*)d_in[10];
  const float* dpar = (const float*)d_in[11];
  const float* opw  = (const float*)d_in[12];
  const float* lavg = (const float*)d_in[13];
  float* out = (float*)d_out;

  char* w = (char*)d_ws;
  auto carve = [&](size_t bytes) -> float* {
    char* p = w;
    w += (bytes + 255) & ~(size_t)255;
    return (float*)p;
  };
  float* XC  = carve((size_t)NB * NL * NDI * 4);
  float* Zb  = carve((size_t)NB * NL * NDI * 4);
  float* XA  = carve((size_t)NB * NL * NDI * 4);
  float* XDB = carve((size_t)16384 * NXDB * 4);
  float* DTb = carve((size_t)16384 * NDI * 4);
  float* Yb  = carve((size_t)16384 * NDI * 4);
  float* AN  = carve((size_t)NDI * NDS * 4);
  float* SSA = carve((size_t)16384 * NDI * NDS * 4);
  float* SSB = carve((size_t)8192  * NDI * NDS * 4);

  k_prep<<<dim3((NDI * NDS + 255) / 256), 256, 0, stream>>>(alog, AN);
  k_inproj<<<dim3((NL + 31) / 32, (2 * NDI) / 256, NB), 128, 0, stream>>>(x, ipw, XC, Zb);
  k_conv<<<dim3(NB * NL), 384, 0, stream>>>(XC, ci, cw, cb, XA);

  const float* prev = nullptr;
  for (int lvl = NDEPTH - 1; lvl >= 0; --lvl) {
    int P = NB << lvl;
    int levoff = NB * ((1 << lvl) - 1);
    int stoff = levoff;
    float* cur = (((NDEPTH - 1 - lvl) & 1) == 0) ? SSA : SSB;
    k_xdb<<<dim3((P + 63) / 64), 128, 0, stream>>>(XA, idxa, xpw, XDB, P, levoff);
    k_dt<<<dim3(NDI / 256, (P + 15) / 16), 128, 0, stream>>>(XDB, dpw, dpb, DTb, P);
    k_ssm<<<dim3(NDI / 256, P), 256, 0, stream>>>(XA, Zb, idxa, sta, XDB, DTb, AN, lavg, dpar,
                                                  prev, cur, Yb, P, levoff, stoff,
                                                  (lvl == NDEPTH - 1) ? 1 : 0);
    k_out<<<dim3(NDM / 256, (P + 31) / 32), 128, 0, stream>>>(Yb, opw, idxa, out, P, levoff);
    prev = cur;
  }
}